// GCMCGraphConv_1975684956870
// MI455X (gfx1250) — compile-verified
//
#include <hip/hip_runtime.h>
#include <hip/hip_bf16.h>

// ---- problem constants (match reference) ----
#define N_SRC     50000
#define N_DST     50000
#define N_EDGES   640000
#define IN_FEATS  256
#define OUT_FEATS 128

typedef __attribute__((ext_vector_type(2))) float v2f;
typedef __attribute__((ext_vector_type(8))) float v8f;

// ------------------------------------------------------------------
// Kernel 0: zero the output accumulator (harness poisons d_out).
// ------------------------------------------------------------------
__global__ void zero_f4(float4* __restrict__ out, int n4) {
    int i = blockIdx.x * blockDim.x + threadIdx.x;
    if (i < n4) out[i] = make_float4(0.f, 0.f, 0.f, 0.f);
}

// ------------------------------------------------------------------
// Kernel 1: h = (feat @ W) * cj   via V_WMMA_F32_16X16X4_F32
//
// One wave32 computes a full 16(M) x 128(N) stripe: 8 accumulators, one per
// 16-col N-tile, all sharing the same A fragment at each K-step.  This cuts
// A (feat) traffic 8x vs one-tile-per-wave: feat is read exactly once
// (51.2 MB @ 23.3 TB/s ~ 2.2us); W (128 KB) stays L2/L0-hot.
//
// fp32 WMMA VGPR layouts (CDNA5 ISA 7.12.2):
//   A 16x4:  lane m = lane&15; VGPR0 holds K = 2*(lane>>4), VGPR1 holds K+1
//   B 4x16:  lane n = lane&15; VGPR0 holds K = 2*(lane>>4), VGPR1 holds K+1
//   C/D 16x16: col n = lane&15; VGPR i holds row m = i + 8*(lane>>4)
// ------------------------------------------------------------------
__global__ void gemm_cj_wmma(const float* __restrict__ feat,
                             const float* __restrict__ W,
                             const float* __restrict__ cj,
                             float* __restrict__ h) {
    const int lane = threadIdx.x & 31;
    const int wave = threadIdx.x >> 5;
    const int mt   = blockIdx.x * (blockDim.x >> 5) + wave;   // M-tile id
    if (mt * 16 >= N_SRC) return;

    const int row0 = mt * 16;
    const int m    = lane & 15;           // A row / B col-within-tile / D col
    const int half = lane >> 4;           // selects K pair & D row half

    const float* __restrict__ arow = feat + (size_t)(row0 + m) * IN_FEATS;

    v8f acc[8];
    #pragma unroll
    for (int t = 0; t < 8; ++t) acc[t] = (v8f){};

    #pragma unroll 2
    for (int k0 = 0; k0 < IN_FEATS; k0 += 4) {
        const int ka = k0 + 2 * half;
        // A fragment: two consecutive K values of this lane's row (8B aligned).
        const v2f a = *(const v2f*)(arow + ka);
        // B fragments: W row-major [K, N]; rows ka, ka+1; col m + 16*t.
        const float* __restrict__ wp = W + (size_t)ka * OUT_FEATS + m;
        #pragma unroll
        for (int t = 0; t < 8; ++t) {
            v2f b;
            b.x = wp[t * 16];
            b.y = wp[OUT_FEATS + t * 16];
            acc[t] = __builtin_amdgcn_wmma_f32_16x16x4_f32(
                         /*neg_a=*/false, a, /*neg_b=*/false, b,
                         /*c_mod=*/(short)0, acc[t],
                         /*reuse_a=*/false, /*reuse_b=*/false);
        }
    }

    // Epilogue: fold in cj[row] once per row, store the 16x128 stripe of h.
    #pragma unroll
    for (int i = 0; i < 8; ++i) {
        const int   row = row0 + i + 8 * half;
        const float s   = cj[row];
        float* __restrict__ hrow = h + (size_t)row * OUT_FEATS + m;
        #pragma unroll
        for (int t = 0; t < 8; ++t) hrow[t * 16] = acc[t][i] * s;
    }
}

// ------------------------------------------------------------------
// Kernel 2: edge scatter-add.  One wave per edge; each lane moves 4 floats
// (32 lanes * 4 = 128 = OUT_FEATS).  ci[dst] folded into the message
// (valid since ci is constant per segment).  h (25.6 MB) and out (25.6 MB)
// both fit in the 192 MB L2, so this phase runs at L2 bandwidth.
// ------------------------------------------------------------------
__global__ void edge_scatter(const float* __restrict__ h,
                             const int* __restrict__ src,
                             const int* __restrict__ dst,
                             const float* __restrict__ ci,
                             float* __restrict__ out) {
    const int lane = threadIdx.x & 31;
    const int e    = (blockIdx.x * blockDim.x + threadIdx.x) >> 5;
    if (e >= N_EDGES) return;

    const int   s     = src[e];      // wave-uniform -> scalar load
    const int   d     = dst[e];
    const float scale = ci[d];

    const float4 v = *(const float4*)(h + (size_t)s * OUT_FEATS + lane * 4);
    float* __restrict__ o = out + (size_t)d * OUT_FEATS + lane * 4;
    atomicAdd(o + 0, v.x * scale);
    atomicAdd(o + 1, v.y * scale);
    atomicAdd(o + 2, v.z * scale);
    atomicAdd(o + 3, v.w * scale);
}

// ------------------------------------------------------------------
extern "C" void kernel_launch(void* const* d_in, const int* in_sizes, int n_in,
                              void* d_out, int out_size, void* d_ws, size_t ws_size,
                              hipStream_t stream) {
    const float* feat   = (const float*)d_in[0];   // [N_SRC, IN_FEATS]
    const float* weight = (const float*)d_in[1];   // [IN_FEATS, OUT_FEATS]
    const float* cj     = (const float*)d_in[2];   // [N_SRC, 1]
    const float* ci     = (const float*)d_in[3];   // [N_DST, 1]
    const int*   src    = (const int*)d_in[4];     // [N_EDGES]
    const int*   dst    = (const int*)d_in[5];     // [N_EDGES]
    float*       out    = (float*)d_out;           // [N_DST, OUT_FEATS]
    float*       h      = (float*)d_ws;            // [N_SRC, OUT_FEATS] scratch

    // 0) zero output
    {
        const int n4 = (N_DST * OUT_FEATS) / 4;    // 1.6M float4
        zero_f4<<<(n4 + 255) / 256, 256, 0, stream>>>((float4*)out, n4);
    }

    // 1) h = (feat @ W) * cj  — 3125 M-stripes, one wave each,
    //    8 waves per 256-thread block -> 391 blocks.
    {
        const int mtiles = N_SRC / 16;             // 3125
        const int wavesPerBlock = 8;
        const int blocks = (mtiles + wavesPerBlock - 1) / wavesPerBlock;
        gemm_cj_wmma<<<blocks, wavesPerBlock * 32, 0, stream>>>(feat, weight, cj, h);
    }

    // 2) scatter-add over edges — one wave per edge, 8 waves per block.
    {
        const int wavesPerBlock = 8;
        const int blocks = (N_EDGES + wavesPerBlock - 1) / wavesPerBlock;
        edge_scatter<<<blocks, wavesPerBlock * 32, 0, stream>>>(h, src, dst, ci, out);
    }
}